// FourierEnhancedAttention_85478439125333
// MI455X (gfx1250) — compile-verified
//
#include <hip/hip_runtime.h>
#include <math.h>

// FEDformer Fourier-Enhanced Attention on gfx1250 (MI455X), wave32 + f32 WMMA.
// All GEMM stages use V_WMMA_F32_16X16X4_F32; one wave per 16x16 output tile.

typedef __attribute__((ext_vector_type(2))) float v2f;
typedef __attribute__((ext_vector_type(8))) float v8f;

#define LSEQ 1024
#define DH   64
#define NM   64
#define TWO_PI 6.28318530717958647692f

__device__ __forceinline__ v8f wmma_f32(v2f a, v2f b, v8f c) {
  // 8 args: (neg_a, A, neg_b, B, c_mod, C, reuse_a, reuse_b)
  return __builtin_amdgcn_wmma_f32_16x16x4_f32(false, a, false, b, (short)0, c,
                                               false, false);
}

// ---------------------------------------------------------------------------
// Fill DFT bases: Fc[t][m]=cos, Fs[t][m]=-sin (forward);
// IB[k][t] = inverse basis rows (k<64: scaled cos, k>=64: scaled -sin),
// scale = (m==0?1:2) / (L * 512 * 512)  (irfft 1/L, Hermitian x2, /d_model^2).
__global__ void k_basis(float* __restrict__ Fc, float* __restrict__ Fs,
                        float* __restrict__ IB) {
  int idx = blockIdx.x * blockDim.x + threadIdx.x;
  if (idx < LSEQ * NM) {
    int t = idx >> 6, m = idx & 63;
    float th = TWO_PI * (float)((t * m) & (LSEQ - 1)) / (float)LSEQ;
    float s, c;
    sincosf(th, &s, &c);
    Fc[t * NM + m] = c;
    Fs[t * NM + m] = -s;
  }
  if (idx < 2 * NM * LSEQ) {
    int k = idx >> 10, t = idx & (LSEQ - 1);
    int m = k & 63;
    float coef = ((m == 0) ? 1.0f : 2.0f) / ((float)LSEQ * 512.0f * 512.0f);
    float th = TWO_PI * (float)((t * m) & (LSEQ - 1)) / (float)LSEQ;
    float s, c;
    sincosf(th, &s, &c);
    IB[k * LSEQ + t] = (k < NM) ? (coef * c) : (-coef * s);
  }
}

// ---------------------------------------------------------------------------
// Forward DFT: spectra[d][m] = sum_t X[b,h,t,d] * basis[t][m].
// One wave -> one 16x16 (d,m) tile; K loop over t (1024, step 4).
__global__ void k_fwd_dft(const float* __restrict__ query,
                          const float* __restrict__ key,
                          const float* __restrict__ Fc,
                          const float* __restrict__ Fs,
                          float* __restrict__ qre, float* __restrict__ qim,
                          float* __restrict__ kre, float* __restrict__ kim) {
  int lane = (int)threadIdx.x;
  int lhalf = lane & 15, hi = lane >> 4;
  int bh = (int)blockIdx.z;
  int src = (int)blockIdx.y;
  int d0 = ((int)blockIdx.x >> 2) * 16;
  int m0 = ((int)blockIdx.x & 3) * 16;
  const float* X = (src == 0 ? query : key) + (size_t)bh * LSEQ * DH;
  float* Ore = (src == 0 ? qre : kre) + (size_t)bh * DH * NM;
  float* Oim = (src == 0 ? qim : kim) + (size_t)bh * DH * NM;
  int d = d0 + lhalf;
  int m = m0 + lhalf;
  v8f accRe = {};
  v8f accIm = {};
  for (int t0 = 0; t0 < LSEQ; t0 += 4) {
    int ta = t0 + 2 * hi;  // this half-wave holds K = ta, ta+1
    v2f a, bc, bs;
    a.x = X[(size_t)ta * DH + d];
    a.y = X[(size_t)(ta + 1) * DH + d];
    bc.x = Fc[ta * NM + m];
    bc.y = Fc[(ta + 1) * NM + m];
    bs.x = Fs[ta * NM + m];
    bs.y = Fs[(ta + 1) * NM + m];
    accRe = wmma_f32(a, bc, accRe);
    accIm = wmma_f32(a, bs, accIm);
  }
  for (int v = 0; v < 8; ++v) {
    int row = d0 + hi * 8 + v;
    Ore[row * NM + m0 + lhalf] = accRe[v];
    Oim[row * NM + m0 + lhalf] = accIm[v];
  }
}

// ---------------------------------------------------------------------------
// scores[q,k] = sum_d qf[d,q]*kf[d,k] (complex, no conj), then complex tanh.
// One wave -> one 16x16 (q,k) tile; K loop over d (64).
__global__ void k_scores_tanh(const float* __restrict__ qre,
                              const float* __restrict__ qim,
                              const float* __restrict__ kre,
                              const float* __restrict__ kim,
                              float* __restrict__ are_,
                              float* __restrict__ aim_) {
  int lane = (int)threadIdx.x;
  int lhalf = lane & 15, hi = lane >> 4;
  int bh = (int)blockIdx.y;
  int q0 = ((int)blockIdx.x >> 2) * 16;
  int k0 = ((int)blockIdx.x & 3) * 16;
  const float* QR = qre + (size_t)bh * DH * NM;
  const float* QI = qim + (size_t)bh * DH * NM;
  const float* KR = kre + (size_t)bh * DH * NM;
  const float* KI = kim + (size_t)bh * DH * NM;
  v8f aRe = {};
  v8f aIm = {};
  for (int dd = 0; dd < DH; dd += 4) {
    int da = dd + 2 * hi;
    v2f aqr, aqi, bkr, bki;
    // A[q, k=d] = qf[d][q]; B[k=d, n=k] = kf[d][k]
    aqr.x = QR[da * NM + q0 + lhalf];
    aqr.y = QR[(da + 1) * NM + q0 + lhalf];
    aqi.x = QI[da * NM + q0 + lhalf];
    aqi.y = QI[(da + 1) * NM + q0 + lhalf];
    bkr.x = KR[da * NM + k0 + lhalf];
    bkr.y = KR[(da + 1) * NM + k0 + lhalf];
    bki.x = KI[da * NM + k0 + lhalf];
    bki.y = KI[(da + 1) * NM + k0 + lhalf];
    aRe = wmma_f32(aqr, bkr, aRe);
    v2f nqi = -aqi;  // f32 WMMA has no A-negate modifier; negate in regs
    aRe = wmma_f32(nqi, bki, aRe);
    aIm = wmma_f32(aqr, bki, aIm);
    aIm = wmma_f32(aqi, bkr, aIm);
  }
  // complex tanh(x+iy) = (sinh 2x + i sin 2y) / (cosh 2x + cos 2y)
  for (int v = 0; v < 8; ++v) {
    float x2 = 2.0f * aRe[v], y2 = 2.0f * aIm[v];
    float re, im;
    if (fabsf(x2) > 30.0f) {
      re = (x2 > 0.0f) ? 1.0f : -1.0f;
      im = 0.0f;
    } else {
      float sy, cy;
      sincosf(y2, &sy, &cy);
      float ep = expf(x2), en = expf(-x2);
      float den = 0.5f * (ep + en) + cy;
      re = 0.5f * (ep - en) / den;
      im = sy / den;
    }
    int row = q0 + hi * 8 + v;
    are_[(size_t)bh * NM * NM + row * NM + k0 + lhalf] = re;
    aim_[(size_t)bh * NM * NM + row * NM + k0 + lhalf] = im;
  }
}

// ---------------------------------------------------------------------------
// ctx[d,q] = sum_k attn[q,k] * kf[d,k] (complex).
// One wave -> one 16x16 (d,q) tile; K loop over k (64).
__global__ void k_context(const float* __restrict__ kre,
                          const float* __restrict__ kim,
                          const float* __restrict__ are_,
                          const float* __restrict__ aim_,
                          float* __restrict__ cre_,
                          float* __restrict__ cim_) {
  int lane = (int)threadIdx.x;
  int lhalf = lane & 15, hi = lane >> 4;
  int bh = (int)blockIdx.y;
  int d0 = ((int)blockIdx.x >> 2) * 16;
  int q0 = ((int)blockIdx.x & 3) * 16;
  const float* KR = kre + (size_t)bh * DH * NM;
  const float* KI = kim + (size_t)bh * DH * NM;
  const float* AR = are_ + (size_t)bh * NM * NM;
  const float* AI = aim_ + (size_t)bh * NM * NM;
  v8f cRe = {};
  v8f cIm = {};
  int dr = d0 + lhalf;
  for (int kk0 = 0; kk0 < NM; kk0 += 4) {
    int ka = kk0 + 2 * hi;
    v2f akr, aki, bar, bai;
    // A[d, k] = kf[d][k] (contiguous in k); B[k, n=q] = attn[q][k]
    akr.x = KR[dr * NM + ka];
    akr.y = KR[dr * NM + ka + 1];
    aki.x = KI[dr * NM + ka];
    aki.y = KI[dr * NM + ka + 1];
    bar.x = AR[(q0 + lhalf) * NM + ka];
    bar.y = AR[(q0 + lhalf) * NM + ka + 1];
    bai.x = AI[(q0 + lhalf) * NM + ka];
    bai.y = AI[(q0 + lhalf) * NM + ka + 1];
    cRe = wmma_f32(akr, bar, cRe);
    v2f nki = -aki;
    cRe = wmma_f32(nki, bai, cRe);
    cIm = wmma_f32(aki, bar, cIm);
    cIm = wmma_f32(akr, bai, cIm);
  }
  for (int v = 0; v < 8; ++v) {
    int row = d0 + hi * 8 + v;
    cre_[(size_t)bh * DH * NM + row * NM + q0 + lhalf] = cRe[v];
    cim_[(size_t)bh * DH * NM + row * NM + q0 + lhalf] = cIm[v];
  }
}

// ---------------------------------------------------------------------------
// Weight mix: mixed[b,h,j,m] = sum_i ctx[b,h,i,m] * (wre + i*wim)[h,i,j,m].
// Writes the [D x 128] packed (re|im) A-matrix for the inverse DFT GEMM.
__global__ void k_mix(const float* __restrict__ cre_,
                      const float* __restrict__ cim_,
                      const float* __restrict__ wre,
                      const float* __restrict__ wim,
                      float* __restrict__ mixed) {
  int m = (int)threadIdx.x;   // 0..63 (contiguous loads across the wave)
  int j = (int)blockIdx.x;    // 0..63
  int bh = (int)blockIdx.y;   // 0..63
  int h = bh & 7;
  const float* CR = cre_ + (size_t)bh * DH * NM;
  const float* CI = cim_ + (size_t)bh * DH * NM;
  float mre = 0.0f, mim = 0.0f;
  for (int i = 0; i < DH; ++i) {
    float cr = CR[i * NM + m];
    float ci = CI[i * NM + m];
    size_t wo = (((size_t)h * DH + i) * DH + j) * NM + m;
    float wr = wre[wo];
    float wi = wim[wo];
    mre = fmaf(cr, wr, fmaf(-ci, wi, mre));
    mim = fmaf(cr, wi, fmaf(ci, wr, mim));
  }
  mixed[((size_t)bh * DH + j) * 128 + m] = mre;
  mixed[((size_t)bh * DH + j) * 128 + 64 + m] = mim;
}

// ---------------------------------------------------------------------------
// Inverse DFT: out[d,t] = sum_{k<128} mixed[d][k] * IB[k][t]  (K=128).
// One wave -> one 16x16 (d,t) tile.
__global__ void k_inv_dft(const float* __restrict__ mixed,
                          const float* __restrict__ IB,
                          float* __restrict__ out) {
  int lane = (int)threadIdx.x;
  int lhalf = lane & 15, hi = lane >> 4;
  int bh = (int)blockIdx.y;
  int tile = (int)blockIdx.x;    // 0..255
  int d0 = (tile >> 6) * 16;     // 4 d-tiles
  int t0 = (tile & 63) * 16;     // 64 t-tiles
  const float* A = mixed + (size_t)bh * DH * 128;
  int d = d0 + lhalf;
  v8f acc = {};
  for (int k0 = 0; k0 < 128; k0 += 4) {
    int ka = k0 + 2 * hi;
    v2f a, b;
    a.x = A[d * 128 + ka];
    a.y = A[d * 128 + ka + 1];
    b.x = IB[ka * LSEQ + t0 + lhalf];
    b.y = IB[(ka + 1) * LSEQ + t0 + lhalf];
    acc = wmma_f32(a, b, acc);
  }
  for (int v = 0; v < 8; ++v) {
    int row = d0 + hi * 8 + v;
    out[((size_t)bh * DH + row) * LSEQ + t0 + lhalf] = acc[v];
  }
}

// ---------------------------------------------------------------------------
extern "C" void kernel_launch(void* const* d_in, const int* in_sizes, int n_in,
                              void* d_out, int out_size, void* d_ws,
                              size_t ws_size, hipStream_t stream) {
  const float* query = (const float*)d_in[0];
  const float* key   = (const float*)d_in[1];
  // d_in[2] = value (unused by the reference forward)
  const float* wre = (const float*)d_in[3];
  const float* wim = (const float*)d_in[4];

  // Workspace layout (floats). Total = 2,883,584 floats = 11.5 MB.
  float* ws = (float*)d_ws;
  float* Fc = ws;               // 65536
  float* Fs = Fc + 65536;       // 65536
  float* IB = Fs + 65536;       // 131072
  float* qre = IB + 131072;     // 262144 each below
  float* qim = qre + 262144;
  float* kre = qim + 262144;
  float* kim = kre + 262144;
  float* atr = kim + 262144;
  float* ati = atr + 262144;
  float* ctr = ati + 262144;
  float* cti = ctr + 262144;
  float* mixed = cti + 262144;  // 524288

  k_basis<<<dim3(512), dim3(256), 0, stream>>>(Fc, Fs, IB);
  k_fwd_dft<<<dim3(16, 2, 64), dim3(32), 0, stream>>>(query, key, Fc, Fs,
                                                      qre, qim, kre, kim);
  k_scores_tanh<<<dim3(16, 64), dim3(32), 0, stream>>>(qre, qim, kre, kim,
                                                       atr, ati);
  k_context<<<dim3(16, 64), dim3(32), 0, stream>>>(kre, kim, atr, ati,
                                                   ctr, cti);
  k_mix<<<dim3(64, 64), dim3(64), 0, stream>>>(ctr, cti, wre, wim, mixed);
  k_inv_dft<<<dim3(256, 64), dim3(32), 0, stream>>>(mixed, IB, (float*)d_out);
}